// Linear_3384434230135
// MI455X (gfx1250) — compile-verified
//
#include <hip/hip_runtime.h>
#include <hip/hip_bf16.h>

// Linear state-space scan, chunk-parallel re-association, fp32 WMMA (CDNA5).
//   x_{k+1} = A x_k + B u_k ;  y_k = C x_k + D u_k
// Batch(64) = M dimension of 16x16 WMMA tiles. Three phases:
//   1) per-chunk local scan with X=0 -> carries v[c]      (128 parallel WGs)
//   2) Q=(A^T)^128 by squaring + sequential carry scan     (1 WG)
//   3) per-chunk replay from correct start state, emit out (128 parallel WGs)

#define NXDIM 64
#define NUDIM 32
#define NYDIM 32
#define BATCHD 64
#define NSTEPS 16384
#define CHUNK_L 128
#define NCHUNK (NSTEPS / CHUNK_L)  // 128

typedef __attribute__((ext_vector_type(2))) float v2f;
typedef __attribute__((ext_vector_type(8))) float v8f;

__device__ __forceinline__ v8f wmma4(v2f a, v2f b, v8f c) {
  // 8-arg pattern: (neg_a, A, neg_b, B, c_mod, C, reuse_a, reuse_b)
  return __builtin_amdgcn_wmma_f32_16x16x4_f32(false, a, false, b, (short)0, c,
                                               false, false);
}

// A-operand (16x4 f32) fragment from row-major LDS [M x ld]:
// lane = lo + 16*hi ; a[j] = X[Mb+lo][kb + 2*hi + j]
__device__ __forceinline__ v2f load_afrag(const float* __restrict__ Xl, int ld,
                                          int Mb, int kb, int lo, int hi) {
  const float* p = Xl + (Mb + lo) * ld + kb + 2 * hi;
  v2f a; a.x = p[0]; a.y = p[1];
  return a;
}

// B-operand (4x16 f32) fragment from row-major LDS [K x ld]:
// lane = lo + 16*hi ; b[j] = W[kb + 2*hi + j][Nb+lo]
__device__ __forceinline__ v2f load_bfrag(const float* __restrict__ Wl, int ld,
                                          int kb, int Nb, int lo, int hi) {
  const float* p = Wl + (kb + 2 * hi) * ld + Nb + lo;
  v2f b; b.x = p[0]; b.y = p[ld];
  return b;
}

// C/D accumulator (16x16 f32, 8 VGPRs): acc[r] = M=Mb+r+8*hi, N=Nb+lo
__device__ __forceinline__ void acc_store_lds(float* __restrict__ Xl, int ld,
                                              int Mb, int Nb, int lo, int hi,
                                              v8f acc) {
#pragma unroll
  for (int r = 0; r < 8; ++r) Xl[(Mb + r + 8 * hi) * ld + Nb + lo] = acc[r];
}

// ---------------------------------------------------------------------------
// Phases 1 & 3: one workgroup per chunk.
//  FULL=false: 512 thr (16 X-waves). X0=0, only the carry v[c] is stored.
//  FULL=true : 768 thr (16 X-waves + 8 Y-waves). X0=s[c]; writes x_full & y.
// ---------------------------------------------------------------------------
template <bool FULL>
__global__ void chunk_scan_kernel(const float* __restrict__ u_in,  // (B,N,NU)
                                  const float* __restrict__ Am,    // (64,64)
                                  const float* __restrict__ Bm,    // (64,32)
                                  const float* __restrict__ Cm,    // (32,64)
                                  const float* __restrict__ Dm,    // (32,32)
                                  const float* __restrict__ s_ws,  // (NCHUNK+1,64,64)
                                  float* __restrict__ v_ws,        // (NCHUNK,64,64)
                                  float* __restrict__ y_out,       // (B,N,NY)
                                  float* __restrict__ x_out) {     // (B,N+1,NX)
  extern __shared__ float smem[];
  float* At = smem;                 // 4096  At[k*64+n] = A[n][k]
  float* Bt = At + 4096;            // 2048  Bt[k*64+n] = B[n][k]
  float* Ct = Bt + 2048;            // 2048  Ct[k*32+n] = C[n][k]
  float* Dt = Ct + 2048;            // 1024  Dt[k*32+n] = D[n][k]
  float* Xbase = Dt + 1024;         // 2 x 4096 state (64x64), ping-pong
  float* Ubase = Xbase + 2 * 4096;  // 2 x 2048 input (64x32), ping-pong

  const int tid = threadIdx.x;
  const int nthr = blockDim.x;
  const int c = blockIdx.x;
  const int lane = tid & 31;
  const int lo = lane & 15;
  const int hi = lane >> 4;
  const int w = tid >> 5;

  // Stage transposed weights into LDS.
  for (int i = tid; i < 64 * 64; i += nthr) {
    int r = i >> 6, cc = i & 63;
    At[cc * 64 + r] = Am[i];
  }
  for (int i = tid; i < 64 * 32; i += nthr) {
    int n = i >> 5, k = i & 31;
    Bt[k * 64 + n] = Bm[i];
  }
  if (FULL) {
    for (int i = tid; i < 32 * 64; i += nthr) {
      int n = i >> 6, k = i & 63;
      Ct[k * 32 + n] = Cm[i];
    }
    for (int i = tid; i < 32 * 32; i += nthr) {
      int n = i >> 5, k = i & 31;
      Dt[k * 32 + n] = Dm[i];
    }
  }

  // Stage X0 (chunk start state) and U_0.
  if (FULL) {
    const float* s0 = s_ws + (size_t)c * 4096;
    for (int i = tid; i < 4096; i += nthr) Xbase[i] = s0[i];
  } else {
    for (int i = tid; i < 4096; i += nthr) Xbase[i] = 0.0f;
  }
  const bool uldr = (tid < BATCHD * (NUDIM / 4));  // 512 float4 loaders
  const int ub = tid >> 3, uq = (tid & 7) * 4;
  {
    const int k0 = c * CHUNK_L;
    if (uldr) {
      float4 uv = *(const float4*)(u_in + ((size_t)ub * NSTEPS + k0) * NUDIM + uq);
      *(float4*)(&Ubase[ub * NUDIM + uq]) = uv;
    }
  }
  __syncthreads();

  const bool isX = (w < 16);
  const int Mb = isX ? ((w >> 2) * 16) : (((w - 16) >> 1) * 16);
  const int Nb = isX ? ((w & 3) * 16) : (((w - 16) & 1) * 16);

  v8f acc = {};  // X-waves: current X_k tile
  if (isX) {
#pragma unroll
    for (int r = 0; r < 8; ++r) acc[r] = Xbase[(Mb + r + 8 * hi) * 64 + Nb + lo];
  }

  int p = 0;
  for (int t = 0; t < CHUNK_L; ++t) {
    const int k = c * CHUNK_L + t;
    const bool havnext = (t + 1 < CHUNK_L);
    float* Xcur = Xbase + p * 4096;
    float* Xnxt = Xbase + (p ^ 1) * 4096;
    float* Ucur = Ubase + p * 2048;
    float* Unxt = Ubase + (p ^ 1) * 2048;

    // Prefetch next U into registers (hides global latency behind WMMA).
    float4 unext;
    if (havnext && uldr)
      unext = *(const float4*)(u_in + ((size_t)ub * NSTEPS + (k + 1)) * NUDIM + uq);

    if (FULL && isX) {  // x_full[k] = X_k (held in acc)
#pragma unroll
      for (int r = 0; r < 8; ++r)
        x_out[((size_t)(Mb + r + 8 * hi) * (NSTEPS + 1) + k) * NXDIM + Nb + lo] =
            acc[r];
    }

    if (isX) {
      // X_{k+1} = X_k * At + U_k * Bt
      v8f nacc = {};
#pragma unroll
      for (int kb = 0; kb < 64; kb += 4)
        nacc = wmma4(load_afrag(Xcur, 64, Mb, kb, lo, hi),
                     load_bfrag(At, 64, kb, Nb, lo, hi), nacc);
#pragma unroll
      for (int kb = 0; kb < 32; kb += 4)
        nacc = wmma4(load_afrag(Ucur, 32, Mb, kb, lo, hi),
                     load_bfrag(Bt, 64, kb, Nb, lo, hi), nacc);
      acc_store_lds(Xnxt, 64, Mb, Nb, lo, hi, nacc);
      acc = nacc;
    } else if (FULL) {
      // Y_k = X_k * Ct + U_k * Dt
      v8f yacc = {};
#pragma unroll
      for (int kb = 0; kb < 64; kb += 4)
        yacc = wmma4(load_afrag(Xcur, 64, Mb, kb, lo, hi),
                     load_bfrag(Ct, 32, kb, Nb, lo, hi), yacc);
#pragma unroll
      for (int kb = 0; kb < 32; kb += 4)
        yacc = wmma4(load_afrag(Ucur, 32, Mb, kb, lo, hi),
                     load_bfrag(Dt, 32, kb, Nb, lo, hi), yacc);
#pragma unroll
      for (int r = 0; r < 8; ++r)
        y_out[((size_t)(Mb + r + 8 * hi) * NSTEPS + k) * NYDIM + Nb + lo] =
            yacc[r];
    }

    if (havnext && uldr) *(float4*)(&Unxt[ub * NUDIM + uq]) = unext;
    __syncthreads();
    p ^= 1;
  }

  if (!FULL && isX) {  // carry v[c] = X_L (row-major)
#pragma unroll
    for (int r = 0; r < 8; ++r)
      v_ws[(size_t)c * 4096 + (Mb + r + 8 * hi) * 64 + Nb + lo] = acc[r];
  }
}

// ---------------------------------------------------------------------------
// Phase 2: single workgroup (512 thr, 16 waves).
//   Q = (A^T)^CHUNK_L via 7 WMMA squarings, then s[c+1] = s[c]*Q + v[c].
// ---------------------------------------------------------------------------
__global__ void carry_scan_kernel(const float* __restrict__ Am,
                                  const float* __restrict__ x0,  // (B,NX)
                                  const float* __restrict__ v_ws,
                                  float* __restrict__ s_ws,
                                  float* __restrict__ x_out) {
  extern __shared__ float smem[];
  float* Qbase = smem;          // 2 x 4096, ping-pong
  float* Sbase = smem + 8192;   // 2 x 4096, ping-pong

  const int tid = threadIdx.x;
  const int nthr = blockDim.x;
  const int lane = tid & 31;
  const int lo = lane & 15;
  const int hi = lane >> 4;
  const int w = tid >> 5;
  const int Mb = (w >> 2) * 16, Nb = (w & 3) * 16;

  for (int i = tid; i < 4096; i += nthr) {
    int r = i >> 6, cc = i & 63;
    Qbase[cc * 64 + r] = Am[i];  // Q = A^T
  }
  for (int i = tid; i < 4096; i += nthr) Sbase[i] = x0[i];
  __syncthreads();

  // Q <- Q^2, seven times  => (A^T)^128
  int q = 0;
  for (int it = 0; it < 7; ++it) {
    float* Qcur = Qbase + q * 4096;
    float* Qnxt = Qbase + (q ^ 1) * 4096;
    v8f acc = {};
#pragma unroll
    for (int kb = 0; kb < 64; kb += 4)
      acc = wmma4(load_afrag(Qcur, 64, Mb, kb, lo, hi),
                  load_bfrag(Qcur, 64, kb, Nb, lo, hi), acc);
    acc_store_lds(Qnxt, 64, Mb, Nb, lo, hi, acc);
    __syncthreads();
    q ^= 1;
  }
  float* Qfin = Qbase + q * 4096;

  // Sequential carry scan over chunks.
  int p = 0;
  for (int c = 0; c < NCHUNK; ++c) {
    float* Scur = Sbase + p * 4096;
    float* Snxt = Sbase + (p ^ 1) * 4096;
    for (int i = tid; i < 4096; i += nthr) s_ws[(size_t)c * 4096 + i] = Scur[i];
    v8f acc;
#pragma unroll
    for (int r = 0; r < 8; ++r)
      acc[r] = v_ws[(size_t)c * 4096 + (Mb + r + 8 * hi) * 64 + Nb + lo];
#pragma unroll
    for (int kb = 0; kb < 64; kb += 4)
      acc = wmma4(load_afrag(Scur, 64, Mb, kb, lo, hi),
                  load_bfrag(Qfin, 64, kb, Nb, lo, hi), acc);
    acc_store_lds(Snxt, 64, Mb, Nb, lo, hi, acc);
    __syncthreads();
    p ^= 1;
  }

  // s[NCHUNK] = x_N : final state -> workspace and x_full[:, N, :]
  float* Sfin = Sbase + p * 4096;
  for (int i = tid; i < 4096; i += nthr) {
    float v = Sfin[i];
    s_ws[(size_t)NCHUNK * 4096 + i] = v;
    int b = i >> 6, col = i & 63;
    x_out[((size_t)b * (NSTEPS + 1) + NSTEPS) * NXDIM + col] = v;
  }
}

extern "C" void kernel_launch(void* const* d_in, const int* in_sizes, int n_in,
                              void* d_out, int out_size, void* d_ws,
                              size_t ws_size, hipStream_t stream) {
  const float* u_in = (const float*)d_in[0];  // d: (B,N,NU,1)
  const float* x0 = (const float*)d_in[1];    // (B,NX,1)
  const float* Am = (const float*)d_in[2];
  const float* Bm = (const float*)d_in[3];
  const float* Cm = (const float*)d_in[4];
  const float* Dm = (const float*)d_in[5];

  float* y_out = (float*)d_out;
  float* x_out = y_out + (size_t)BATCHD * NSTEPS * NYDIM;

  float* v_ws = (float*)d_ws;                            // NCHUNK * 4096 f32
  float* s_ws = v_ws + (size_t)NCHUNK * BATCHD * NXDIM;  // (NCHUNK+1) * 4096 f32

  const size_t chunk_smem = 21504 * sizeof(float);  // 84 KB
  const size_t carry_smem = 16384 * sizeof(float);  // 64 KB

  chunk_scan_kernel<false><<<NCHUNK, 512, chunk_smem, stream>>>(
      u_in, Am, Bm, Cm, Dm, nullptr, v_ws, nullptr, nullptr);
  carry_scan_kernel<<<1, 512, carry_smem, stream>>>(Am, x0, v_ws, s_ws, x_out);
  chunk_scan_kernel<true><<<NCHUNK, 768, chunk_smem, stream>>>(
      u_in, Am, Bm, Cm, Dm, s_ws, nullptr, y_out, x_out);
}